// GCN_48112223650068
// MI455X (gfx1250) — compile-verified
//
#include <hip/hip_runtime.h>

#define NN 100000
#define NE 3200000
#define DD 256
#define FACTOR 0.5f

typedef __attribute__((ext_vector_type(2))) float v2f;
typedef __attribute__((ext_vector_type(8))) float v8f;

// ---------------------------------------------------------------------------
// SpMM: out[rows[e], :] += (vals[e]*FACTOR) * h[cols[e], :]
// One wave per edge; 32 lanes x 8 floats = 256-wide row.
// Edge lists streamed once -> non-temporal loads (preserve L2 for features).
// Scatter via hardware GLOBAL_ATOMIC_ADD_F32.
// ---------------------------------------------------------------------------
__global__ __launch_bounds__(256) void gcn_spmm_kernel(
    const float* __restrict__ h,
    const float* __restrict__ vals,
    const int*   __restrict__ rows,
    const int*   __restrict__ cols,
    float* __restrict__ out)
{
    const int lane   = threadIdx.x & 31;
    const int wave   = (blockIdx.x * blockDim.x + threadIdx.x) >> 5;
    const int nwaves = (gridDim.x * blockDim.x) >> 5;

    for (int e = wave; e < NE; e += nwaves) {
        const int   r = __builtin_nontemporal_load(rows + e);
        const int   c = __builtin_nontemporal_load(cols + e);
        const float w = __builtin_nontemporal_load(vals + e) * FACTOR;

        const float4* src = (const float4*)(h + (size_t)c * DD);
        float*        dst = out + (size_t)r * DD + lane * 8;

        float4 a = src[lane * 2 + 0];
        float4 b = src[lane * 2 + 1];

        unsafeAtomicAdd(dst + 0, w * a.x);
        unsafeAtomicAdd(dst + 1, w * a.y);
        unsafeAtomicAdd(dst + 2, w * a.z);
        unsafeAtomicAdd(dst + 3, w * a.w);
        unsafeAtomicAdd(dst + 4, w * b.x);
        unsafeAtomicAdd(dst + 5, w * b.y);
        unsafeAtomicAdd(dst + 6, w * b.z);
        unsafeAtomicAdd(dst + 7, w * b.w);
    }
}

// ---------------------------------------------------------------------------
// GEMM: C[M,256] = act(A[M,256] @ W[256,256]) via V_WMMA_F32_16X16X4_F32.
// Block = 512 threads (16 waves) owns one 16-row strip of A:
//   - A strip (16x256 f32 = 16 KB) staged to LDS exactly once (global A read 1x)
//   - wave w computes the 16x16 tile for column tile w; B (=W) stays L2-hot
// LDS row stride padded to 260 floats: stride mod 64 banks == 4, so the 64
// dwords A[m][kk..kk+3] (m=0..15) of each fragment fetch map to 64 distinct
// banks -> conflict-free DS_LOAD_B64 fragment reads.
// M = 100000 = 6250*16 exactly -> no tail, EXEC all-ones for WMMA.
// ---------------------------------------------------------------------------
#define LDS_STRIDE 260

template <bool RELU>
__global__ __launch_bounds__(512) void gcn_gemm_kernel(
    const float* __restrict__ A,   // [M, 256]
    const float* __restrict__ W,   // [256, 256] row-major
    float* __restrict__ C)         // [M, 256]
{
    __shared__ float As[16 * LDS_STRIDE];

    const int t       = threadIdx.x;
    const int lane    = t & 31;
    const int wave    = t >> 5;          // 0..15 -> column tile
    const int rowTile = blockIdx.x;      // 0..6249

    // Stage the 16x256 A strip (1024 float4) into padded LDS.
    const float4* Ag = (const float4*)(A + (size_t)rowTile * 16 * DD);
#pragma unroll
    for (int i = 0; i < 2; ++i) {
        const int idx = t + i * 512;     // 0..1023
        const int r   = idx >> 6;        // 64 float4 per row
        const int c4  = idx & 63;
        *(float4*)(&As[r * LDS_STRIDE + c4 * 4]) = Ag[idx];
    }
    __syncthreads();

    const int mn   = lane & 15;
    const int n    = wave * 16 + mn;            // B/C column for this lane
    const int koff = (lane < 16) ? 0 : 2;       // K-pair select per half-wave
    const float* Alds = As + mn * LDS_STRIDE;   // A row for this lane

    v8f acc = {};
#pragma unroll 4
    for (int kk = 0; kk < DD; kk += 4) {
        const int k0 = kk + koff;               // even -> 8B-aligned pair
        v2f a = *(const v2f*)(Alds + k0);       // ds_load_b64, conflict-free
        v2f b;
        b.x = W[(size_t)k0 * DD + n];
        b.y = W[(size_t)(k0 + 1) * DD + n];
        acc = __builtin_amdgcn_wmma_f32_16x16x4_f32(
            /*neg_a=*/false, a, /*neg_b=*/false, b,
            /*c_mod=*/(short)0, acc, /*reuse_a=*/false, /*reuse_b=*/false);
    }

    // C/D layout: VGPR r -> M = r (lanes 0-15) or 8+r (lanes 16-31), N = lane%16
    const int mBase = rowTile * 16 + ((lane < 16) ? 0 : 8);
#pragma unroll
    for (int r = 0; r < 8; ++r) {
        float v = acc[r];
        if (RELU) v = fmaxf(v, 0.0f);
        C[(size_t)(mBase + r) * DD + n] = v;
    }
}

// ---------------------------------------------------------------------------
// Pipeline: h = x
//   s1 = spmm(x)*.5 -> ws ; s2 = spmm(s1)*.5 -> out
//   h1 = relu(out @ W0)   -> ws
//   s3 = spmm(ws)*.5 -> out ; s4 = spmm(out)*.5 -> ws
//   out = ws @ W1
// Needs one N*D f32 scratch buffer (102.4 MB) in d_ws.
// ---------------------------------------------------------------------------
extern "C" void kernel_launch(void* const* d_in, const int* in_sizes, int n_in,
                              void* d_out, int out_size, void* d_ws, size_t ws_size,
                              hipStream_t stream) {
    const float* x    = (const float*)d_in[0];
    const float* vals = (const float*)d_in[1];
    const float* W0   = (const float*)d_in[2];
    const float* W1   = (const float*)d_in[3];
    const int*   rows = (const int*)d_in[4];
    const int*   cols = (const int*)d_in[5];

    float* out = (float*)d_out;
    float* ws  = (float*)d_ws;

    const size_t featBytes = (size_t)NN * DD * sizeof(float);
    const int spmmBlocks = 8192;        // 65536 waves, ~49 edges each
    const int gemmBlocks = NN / 16;     // 6250 row strips, 16 waves each

    // s1 = spmm(x) -> ws
    hipMemsetAsync(ws, 0, featBytes, stream);
    gcn_spmm_kernel<<<spmmBlocks, 256, 0, stream>>>(x, vals, rows, cols, ws);
    // s2 = spmm(ws) -> out
    hipMemsetAsync(out, 0, featBytes, stream);
    gcn_spmm_kernel<<<spmmBlocks, 256, 0, stream>>>(ws, vals, rows, cols, out);
    // h1 = relu(out @ W0) -> ws
    gcn_gemm_kernel<true><<<gemmBlocks, 512, 0, stream>>>(out, W0, ws);
    // s3 = spmm(ws) -> out
    hipMemsetAsync(out, 0, featBytes, stream);
    gcn_spmm_kernel<<<spmmBlocks, 256, 0, stream>>>(ws, vals, rows, cols, out);
    // s4 = spmm(out) -> ws
    hipMemsetAsync(ws, 0, featBytes, stream);
    gcn_spmm_kernel<<<spmmBlocks, 256, 0, stream>>>(out, vals, rows, cols, ws);
    // out = ws @ W1
    gcn_gemm_kernel<false><<<gemmBlocks, 512, 0, stream>>>(ws, W1, out);
}